// MaskedMSELoss_48009144435087
// MI455X (gfx1250) — compile-verified
//
#include <hip/hip_runtime.h>

typedef __attribute__((ext_vector_type(2))) float v2f;
typedef __attribute__((ext_vector_type(8))) float v8f;

__global__ void MaskedMSE_zero_kernel(float* out) {
    if (threadIdx.x == 0) out[0] = 0.0f;
}

// One wave32 per row. Per iteration: two 128-element sub-chunks.
// Rank computation via ballot+popcount (no LDS crossbar, scalar running count).
// Gathers are unconditional at clip(rank,0) (same cachelines as masked reads),
// mask applied branchlessly. Final wave reduction via V_WMMA_F32_16X16X4_F32.
__global__ __launch_bounds__(256) void MaskedMSE_row_kernel(
    const float* __restrict__ pred,
    const float* __restrict__ tgt,
    const int*   __restrict__ mask,
    float* __restrict__ out,
    int B, int S)
{
    const int lane = threadIdx.x & 31;
    const int wave = threadIdx.x >> 5;
    const int row  = blockIdx.x * 8 + wave;
    if (row >= B) return;  // uniform per wave

    const size_t rowOff  = (size_t)row * (size_t)S;
    const float* predRow = pred + rowOff;
    const float* tgtRow  = tgt  + rowOff;
    const int*   maskRow = mask + rowOff;

    const unsigned lt = (1u << lane) - 1u;   // lanemask_lt for wave32

    float acc = 0.0f;
    int runningCount = 0;   // masked positions consumed so far in this row

    for (int chunk = 0; chunk < S; chunk += 256) {
        const int jA = chunk + lane * 4;
        const int jB = jA + 128;
        // Issue all four b128 streaming loads up front.
        const int4   mA = *(const int4*)(maskRow + jA);
        const float4 pA = *(const float4*)(predRow + jA);
        const int4   mB = *(const int4*)(maskRow + jB);
        const float4 pB = *(const float4*)(predRow + jB);

        // ---------------- sub-chunk A ----------------
        {
            const int b0 = (mA.x != 0), b1 = (mA.y != 0), b2 = (mA.z != 0), b3 = (mA.w != 0);
            const unsigned W0 = __builtin_amdgcn_ballot_w32(b0);
            const unsigned W1 = __builtin_amdgcn_ballot_w32(b1);
            const unsigned W2 = __builtin_amdgcn_ballot_w32(b2);
            const unsigned W3 = __builtin_amdgcn_ballot_w32(b3);
            // local inclusive counts within this lane's 4 elements
            const int s0 = b0, s1 = s0 + b1, s2 = s1 + b2, s3 = s2 + b3;
            // exclusive prefix across lanes: 4 VALU popcounts, no shuffles
            const int excl = __popc(W0 & lt) + __popc(W1 & lt)
                           + __popc(W2 & lt) + __popc(W3 & lt);
            const int base = runningCount + excl;

            const int r0 = max(base + s0 - 1, 0);
            const int r1 = max(base + s1 - 1, 0);
            const int r2 = max(base + s2 - 1, 0);
            const int r3 = max(base + s3 - 1, 0);
            // 4 independent unconditional gathers -> single wait, full MLP
            const float t0 = tgtRow[r0];
            const float t1 = tgtRow[r1];
            const float t2 = tgtRow[r2];
            const float t3 = tgtRow[r3];

            const float d0 = b0 ? (pA.x - t0) : 0.0f;
            const float d1 = b1 ? (pA.y - t1) : 0.0f;
            const float d2 = b2 ? (pA.z - t2) : 0.0f;
            const float d3 = b3 ? (pA.w - t3) : 0.0f;
            acc = fmaf(d0, d0, acc);
            acc = fmaf(d1, d1, acc);
            acc = fmaf(d2, d2, acc);
            acc = fmaf(d3, d3, acc);

            // wave total: pure SALU (s_bcnt) -> cheap carried dependency
            runningCount += __popc(W0) + __popc(W1) + __popc(W2) + __popc(W3);
        }

        // ---------------- sub-chunk B ----------------
        {
            const int b0 = (mB.x != 0), b1 = (mB.y != 0), b2 = (mB.z != 0), b3 = (mB.w != 0);
            const unsigned W0 = __builtin_amdgcn_ballot_w32(b0);
            const unsigned W1 = __builtin_amdgcn_ballot_w32(b1);
            const unsigned W2 = __builtin_amdgcn_ballot_w32(b2);
            const unsigned W3 = __builtin_amdgcn_ballot_w32(b3);
            const int s0 = b0, s1 = s0 + b1, s2 = s1 + b2, s3 = s2 + b3;
            const int excl = __popc(W0 & lt) + __popc(W1 & lt)
                           + __popc(W2 & lt) + __popc(W3 & lt);
            const int base = runningCount + excl;

            const int r0 = max(base + s0 - 1, 0);
            const int r1 = max(base + s1 - 1, 0);
            const int r2 = max(base + s2 - 1, 0);
            const int r3 = max(base + s3 - 1, 0);
            const float t0 = tgtRow[r0];
            const float t1 = tgtRow[r1];
            const float t2 = tgtRow[r2];
            const float t3 = tgtRow[r3];

            const float d0 = b0 ? (pB.x - t0) : 0.0f;
            const float d1 = b1 ? (pB.y - t1) : 0.0f;
            const float d2 = b2 ? (pB.z - t2) : 0.0f;
            const float d3 = b3 ? (pB.w - t3) : 0.0f;
            acc = fmaf(d0, d0, acc);
            acc = fmaf(d1, d1, acc);
            acc = fmaf(d2, d2, acc);
            acc = fmaf(d3, d3, acc);

            runningCount += __popc(W0) + __popc(W1) + __popc(W2) + __popc(W3);
        }
    }

    // ---- Wave reduction via V_WMMA_F32_16X16X4_F32 "multiply by ones" ----
    // A (16x4 f32): VGPR0 = acc (lanes 0-15 -> K=0, lanes 16-31 -> K=2), VGPR1 = 0.
    // B (4x16) = all ones  =>  D[m][n] = acc[m] + acc[m+16]  (exact f32, RNE).
    v2f a; a[0] = acc;  a[1] = 0.0f;
    v2f b; b[0] = 1.0f; b[1] = 1.0f;
    v8f c = {};
    v8f d = __builtin_amdgcn_wmma_f32_16x16x4_f32(
        /*neg_a=*/false, a, /*neg_b=*/false, b,
        /*c_mod=*/(short)0, c, /*reuse_a=*/false, /*reuse_b=*/false);

    // Lanes 0-15 hold D rows M=0..7 in VGPRs 0..7; lanes 16-31 hold M=8..15.
    float s = ((d[0] + d[1]) + (d[2] + d[3])) + ((d[4] + d[5]) + (d[6] + d[7]));
    s += __shfl_xor(s, 16, 32);   // combine the two M halves -> full wave sum
    if (lane == 0) atomicAdd(out, s);
}

extern "C" void kernel_launch(void* const* d_in, const int* in_sizes, int n_in,
                              void* d_out, int out_size, void* d_ws, size_t ws_size,
                              hipStream_t stream) {
    const float* pred = (const float*)d_in[0];
    const float* tgt  = (const float*)d_in[1];
    const int*   mask = (const int*)d_in[2];
    float* out = (float*)d_out;

    const int S = 4096;
    const int B = in_sizes[0] / S;

    MaskedMSE_zero_kernel<<<1, 32, 0, stream>>>(out);

    const int rowsPerBlock = 8;                       // 8 waves * wave32 = 256 threads
    const int grid = (B + rowsPerBlock - 1) / rowsPerBlock;
    MaskedMSE_row_kernel<<<grid, 256, 0, stream>>>(pred, tgt, mask, out, B, S);
}